// CapsuleLayer_35502199669310
// MI455X (gfx1250) — compile-verified
//
#include <hip/hip_runtime.h>

// CapsuleLayer dynamic routing, fp32, CDNA5 (gfx1250, wave32).
// ws layout: u_hat [B,N,I,D] 256MB | b_logits [B,N,I] 8MB | s [B,N,D] | v [B,N,D]

typedef __attribute__((ext_vector_type(2))) float v2f;
typedef __attribute__((ext_vector_type(8))) float v8f;

#define B_ 16
#define N_ 64
#define I_ 2048
#define D_ 32
#define E_ 16

// ---------------------------------------------------------------------------
// K1: u_hat[b,n,i,d] = sum_e W[n,i,d,e] * x[b,i,e]  via V_WMMA_F32_16X16X4_F32
//     Also accumulates s0[b,n,d] = sum_i u_hat[b,n,i,d] (iter-0 uniform c).
// One wave per (n,i): A = W-slice [16x4 f32 tiles], B = x-slice [4x16], two
// d-halves, K=16 chained as 4 wmma steps. Grid: (N, I/64), 8 waves x 8 i's.
// ---------------------------------------------------------------------------
__global__ __launch_bounds__(256) void capsule_uhat_wmma(
    const float* __restrict__ x,      // [B,I,E]
    const float* __restrict__ W,      // [N,I,D,E]
    float* __restrict__ u_hat,        // [B,N,I,D]
    float* __restrict__ s_acc)        // [B,N,D] (pre-zeroed)
{
    __shared__ float sacc[D_][B_];
    const int tid  = threadIdx.x;
    const int lane = tid & 31;
    const int wave = tid >> 5;
    const int n    = blockIdx.x;
    const int i0   = blockIdx.y * 64 + wave * 8;

    for (int t = tid; t < D_ * B_; t += 256) ((float*)sacc)[t] = 0.f;
    __syncthreads();

    const int lo = lane & 15;   // A: row M (=d-local); B/C: col N (=b)
    const int hi = lane >> 4;   // K half select

    float racc[2][8];
#pragma unroll
    for (int h = 0; h < 2; ++h)
#pragma unroll
        for (int j = 0; j < 8; ++j) racc[h][j] = 0.f;

    for (int t = 0; t < 8; ++t) {
        const int i = i0 + t;
        // B operand: B[K=e][N=b] = x[b,i,e]; lane holds (K=4kk+2hi, K+1) at col lo
        v2f bt[4];
        const float* xp = x + ((size_t)lo * I_ + i) * E_ + 2 * hi;
#pragma unroll
        for (int kk = 0; kk < 4; ++kk) {
            bt[kk].x = xp[4 * kk + 0];
            bt[kk].y = xp[4 * kk + 1];
        }
#pragma unroll
        for (int h = 0; h < 2; ++h) {
            const int dbase = h * 16;
            // A operand: A[M=d-local][K=e-local] = W[n,i,dbase+M,4kk+K]
            const float* wp =
                W + (((size_t)n * I_ + i) * D_ + dbase + lo) * E_ + 2 * hi;
            v8f c = {};
#pragma unroll
            for (int kk = 0; kk < 4; ++kk) {
                v2f a;
                a.x = wp[4 * kk + 0];
                a.y = wp[4 * kk + 1];
                c = __builtin_amdgcn_wmma_f32_16x16x4_f32(
                        false, a, false, bt[kk], (short)0, c, false, false);
            }
            // C/D layout: vgpr j, lane -> M = j + 8*hi, N = lo
            const int b  = lo;
            const int d0 = dbase + 8 * hi;           // 8 consecutive d per lane
            float* up = u_hat + (((size_t)b * N_ + n) * I_ + i) * D_ + d0;
#pragma unroll
            for (int j = 0; j < 8; ++j) {
                up[j] = c[j];
                racc[h][j] += c[j];
            }
        }
    }
    // fold per-wave i-sums into LDS, then one atomic set per workgroup
#pragma unroll
    for (int h = 0; h < 2; ++h) {
        const int d0 = h * 16 + 8 * hi;
#pragma unroll
        for (int j = 0; j < 8; ++j) atomicAdd(&sacc[d0 + j][lo], racc[h][j]);
    }
    __syncthreads();
    for (int t = tid; t < D_ * B_; t += 256) {
        int d = t / B_, b = t % B_;
        atomicAdd(&s_acc[((size_t)b * N_ + n) * D_ + d], sacc[d][b]);
    }
}

// ---------------------------------------------------------------------------
// K2: squash over D. One wave per (b,n); lane = d. v = s2/(1+s2) * s/sqrt(s2+eps)
// ---------------------------------------------------------------------------
__global__ __launch_bounds__(256) void capsule_squash(
    const float* __restrict__ s, float* __restrict__ v, float pre_scale)
{
    const int wg   = (blockIdx.x * 256 + threadIdx.x) >> 5;  // (b*N+n)
    const int lane = threadIdx.x & 31;
    const size_t idx = (size_t)wg * D_ + lane;
    float val = s[idx] * pre_scale;
    float s2  = val * val;
#pragma unroll
    for (int off = 16; off; off >>= 1) s2 += __shfl_xor(s2, off);
    float sc = (s2 / (1.f + s2)) * rsqrtf(s2 + 1e-7f);
    v[idx] = sc * val;
}

// ---------------------------------------------------------------------------
// K3: fused routing boundary (one read of u_hat):
//   b_new[b,n,i] = b_prev + sum_d v[b,n,d]*u_hat[b,n,i,d]
//   c = softmax_n(b_new);  s_acc[b,n,d] += sum_i c*u_hat[b,n,i,d]
// Lane owns n=lane and n=lane+32 (64-wide softmax = 2 vals + wave reduce).
// Grid: (B, I/64); 8 waves x 8 i's. LDS reduce then global atomics for s.
// ---------------------------------------------------------------------------
__global__ __launch_bounds__(256) void capsule_route(
    const float* __restrict__ u_hat,   // [B,N,I,D]
    const float* __restrict__ v_in,    // [B,N,D]
    float* __restrict__ b_logits,      // [B,N,I] in/out (in-place, lane-owned)
    float* __restrict__ s_acc,         // [B,N,D] (pre-zeroed)
    int use_prev_b)
{
    __shared__ float red[N_][D_];
    const int lane  = threadIdx.x & 31;
    const int wave  = threadIdx.x >> 5;
    const int b     = blockIdx.x;
    const int ibase = blockIdx.y * 64 + wave * 8;
    const int n0 = lane, n1 = lane + 32;

    for (int t = threadIdx.x; t < N_ * D_; t += 256) ((float*)red)[t] = 0.f;

    float v0[D_], v1[D_], acc0[D_], acc1[D_];
    {
        const float* p0 = v_in + ((size_t)b * N_ + n0) * D_;
        const float* p1 = v_in + ((size_t)b * N_ + n1) * D_;
#pragma unroll
        for (int d = 0; d < D_; ++d) {
            v0[d] = p0[d]; v1[d] = p1[d];
            acc0[d] = 0.f; acc1[d] = 0.f;
        }
    }
    __syncthreads();

    for (int j = 0; j < 8; ++j) {
        const int i = ibase + j;
        const float* u0p = u_hat + (((size_t)b * N_ + n0) * I_ + i) * D_;
        const float* u1p = u_hat + (((size_t)b * N_ + n1) * I_ + i) * D_;
        float dot0 = 0.f, dot1 = 0.f;
#pragma unroll
        for (int d = 0; d < D_; ++d) {
            dot0 += v0[d] * u0p[d];
            dot1 += v1[d] * u1p[d];
        }
        const size_t bi0 = ((size_t)b * N_ + n0) * I_ + i;
        const size_t bi1 = ((size_t)b * N_ + n1) * I_ + i;
        float l0 = dot0, l1 = dot1;
        if (use_prev_b) { l0 += b_logits[bi0]; l1 += b_logits[bi1]; }
        b_logits[bi0] = l0;
        b_logits[bi1] = l1;
        // softmax over the 64 capsules (2 per lane + wave32 reduction)
        float m = fmaxf(l0, l1);
#pragma unroll
        for (int off = 16; off; off >>= 1) m = fmaxf(m, __shfl_xor(m, off));
        float e0 = __expf(l0 - m), e1 = __expf(l1 - m);
        float ss = e0 + e1;
#pragma unroll
        for (int off = 16; off; off >>= 1) ss += __shfl_xor(ss, off);
        float inv = 1.f / ss;
        float c0 = e0 * inv, c1 = e1 * inv;
        // weighted accumulate (u re-read hits WGP$/L2 — just loaded above)
#pragma unroll
        for (int d = 0; d < D_; ++d) {
            acc0[d] += c0 * u0p[d];
            acc1[d] += c1 * u1p[d];
        }
    }
#pragma unroll
    for (int d = 0; d < D_; ++d) {
        atomicAdd(&red[n0][d], acc0[d]);
        atomicAdd(&red[n1][d], acc1[d]);
    }
    __syncthreads();
    for (int t = threadIdx.x; t < N_ * D_; t += 256) {
        int n = t / D_, d = t % D_;
        atomicAdd(&s_acc[((size_t)b * N_ + n) * D_ + d], red[n][d]);
    }
}

// ---------------------------------------------------------------------------
extern "C" void kernel_launch(void* const* d_in, const int* in_sizes, int n_in,
                              void* d_out, int out_size, void* d_ws, size_t ws_size,
                              hipStream_t stream)
{
    const float* x = (const float*)d_in[0];   // [16,2048,16]
    const float* W = (const float*)d_in[1];   // [64,2048,32,16]
    float* out = (float*)d_out;               // [16,64,32]

    char* ws = (char*)d_ws;
    float* u_hat = (float*)ws;
    size_t off = (size_t)B_ * N_ * I_ * D_ * sizeof(float);      // 256 MB
    float* blog = (float*)(ws + off);
    off += (size_t)B_ * N_ * I_ * sizeof(float);                 // 8 MB
    float* sbuf = (float*)(ws + off);
    off += (size_t)B_ * N_ * D_ * sizeof(float);
    float* vbuf = (float*)(ws + off);

    const size_t sbytes = (size_t)B_ * N_ * D_ * sizeof(float);
    const dim3 g1(N_, I_ / 64);   // 2048 WGs
    const dim3 g3(B_, I_ / 64);   // 512 WGs
    const int  gsq = (B_ * N_ * 32) / 256;

    // iter 0: uniform c = 1/64 folded into K1's s0 accumulation
    hipMemsetAsync(sbuf, 0, sbytes, stream);
    capsule_uhat_wmma<<<g1, 256, 0, stream>>>(x, W, u_hat, sbuf);
    capsule_squash<<<gsq, 256, 0, stream>>>(sbuf, vbuf, 1.0f / N_);   // v0

    // iter 1: b1 = v0.u ; c1 = softmax(b1) ; s1
    hipMemsetAsync(sbuf, 0, sbytes, stream);
    capsule_route<<<g3, 256, 0, stream>>>(u_hat, vbuf, blog, sbuf, 0);
    capsule_squash<<<gsq, 256, 0, stream>>>(sbuf, vbuf, 1.0f);        // v1

    // iter 2: b2 = b1 + v1.u ; c2 = softmax(b2) ; s2 ; output = squash(s2)
    hipMemsetAsync(sbuf, 0, sbytes, stream);
    capsule_route<<<g3, 256, 0, stream>>>(u_hat, vbuf, blog, sbuf, 1);
    capsule_squash<<<gsq, 256, 0, stream>>>(sbuf, out, 1.0f);
}